// NTXentLoss_63264868270602
// MI455X (gfx1250) — compile-verified
//
#include <hip/hip_runtime.h>
#include <math.h>

typedef __attribute__((ext_vector_type(2))) float v2f;
typedef __attribute__((ext_vector_type(8))) float v8f;

#define BN 4096
#define DN 256
#define NN 8192
#define INV_T 14.2857142857142857f /* 1/0.07 */
#define TILE 128 /* square block tile */
#define KC 32    /* K chunk staged in LDS */
#define LDSD 34  /* padded LDS row stride (floats): bank = (2m+k)%64 distinct per lane */

__device__ __forceinline__ const float* row_ptr(const float* zi, const float* zj, int row) {
  return (row < BN) ? (zi + (size_t)row * DN) : (zj + (size_t)(row - BN) * DN);
}

// ---------------- Kernel 1: inverse norms + zero accumulators ----------------
__global__ __launch_bounds__(256) void ntx_norm_kernel(const float* __restrict__ zi,
                                                       const float* __restrict__ zj,
                                                       float* __restrict__ inv_norm,
                                                       float* __restrict__ rowsum,
                                                       float* __restrict__ pos_acc) {
  const int row = blockIdx.x;
  const int t = threadIdx.x;
  const float* p = row_ptr(zi, zj, row);
  float v = p[t];
  float s = v * v;
#pragma unroll
  for (int m = 16; m >= 1; m >>= 1) s += __shfl_xor(s, m, 32);
  __shared__ float red[8];
  if ((t & 31) == 0) red[t >> 5] = s;
  __syncthreads();
  if (t == 0) {
    float tot = 0.f;
#pragma unroll
    for (int i = 0; i < 8; ++i) tot += red[i];
    inv_norm[row] = 1.0f / fmaxf(sqrtf(tot), 1e-12f);
    rowsum[row] = 0.0f;
    if (row == 0) *pos_acc = 0.0f;
  }
}

// ---------------- Kernel 2: upper-triangular tiled Gram + exp + row sums ----------------
__global__ __launch_bounds__(256) void ntx_gram_kernel(const float* __restrict__ zi,
                                                       const float* __restrict__ zj,
                                                       const float* __restrict__ inv_norm,
                                                       float* __restrict__ rowsum,
                                                       float* __restrict__ pos_acc) {
  const int bj = blockIdx.x;  // 128-col tile
  const int bi = blockIdx.y;  // 128-row tile
  if (bj < bi) return;        // block-uniform exit: EXEC stays all-1 for WMMA
  const int Abase = bi * TILE;
  const int Bbase = bj * TILE;
  const bool upper = (bj > bi);

  __shared__ float As[TILE * LDSD];
  __shared__ float Bs[TILE * LDSD];

  const int t = threadIdx.x;
  const int lane = t & 31;
  const int wave = t >> 5;
  const int wr = wave & 3;             // wave row group: rows [32*wr, 32*wr+32)
  const int wc = wave >> 2;            // wave col group: cols [64*wc, 64*wc+64)
  const int mi = lane & 15;            // M for A-frags / N for B-frags
  const int koffL = (lane >> 4) << 1;  // 0 (lanes 0-15) or 2 (lanes 16-31)

  v8f acc[2][4] = {};  // 2 row-subtiles x 4 col-subtiles of 16x16

  // LDS fragment base pointers (per-wave constant; LDS refilled in place per chunk)
  const float* aRow[2];
  const float* bRow[4];
#pragma unroll
  for (int i = 0; i < 2; ++i) aRow[i] = &As[(wr * 32 + i * 16 + mi) * LDSD + koffL];
#pragma unroll
  for (int j = 0; j < 4; ++j) bRow[j] = &Bs[(wc * 64 + j * 16 + mi) * LDSD + koffL];

  const int cshift = (t & 7) * 4;  // staging: 4-float column chunk within KC
  const int rgrp = t >> 3;         // staging: base row (0..31)

  for (int kc0 = 0; kc0 < DN; kc0 += KC) {
    __syncthreads();
// cooperative staging: 128 rows x 32 cols per matrix, normalized on the fly
#pragma unroll
    for (int p = 0; p < 4; ++p) {
      const int r = rgrp + 32 * p;
      const int gra = Abase + r;
      const float4 va = *(const float4*)(row_ptr(zi, zj, gra) + kc0 + cshift);
      const float sa = inv_norm[gra];
      As[r * LDSD + cshift + 0] = va.x * sa;
      As[r * LDSD + cshift + 1] = va.y * sa;
      As[r * LDSD + cshift + 2] = va.z * sa;
      As[r * LDSD + cshift + 3] = va.w * sa;
      const int grb = Bbase + r;
      const float4 vb = *(const float4*)(row_ptr(zi, zj, grb) + kc0 + cshift);
      const float sb = inv_norm[grb];
      Bs[r * LDSD + cshift + 0] = vb.x * sb;
      Bs[r * LDSD + cshift + 1] = vb.y * sb;
      Bs[r * LDSD + cshift + 2] = vb.z * sb;
      Bs[r * LDSD + cshift + 3] = vb.w * sb;
    }
    __syncthreads();

#pragma unroll
    for (int k = 0; k < KC; k += 4) {
      v2f a[2], b[4];
#pragma unroll
      for (int i = 0; i < 2; ++i) a[i] = *(const v2f*)(aRow[i] + k);
#pragma unroll
      for (int j = 0; j < 4; ++j) b[j] = *(const v2f*)(bRow[j] + k);
#pragma unroll
      for (int i = 0; i < 2; ++i)
#pragma unroll
        for (int j = 0; j < 4; ++j)
          acc[i][j] = __builtin_amdgcn_wmma_f32_16x16x4_f32(false, a[i], false, b[j],
                                                            (short)0, acc[i][j], false,
                                                            false);
    }
  }

  // Epilogue. C/D layout: VGPR r, lanes 0-15: M=r, N=lane; lanes 16-31: M=r+8, N=lane-16
  const int half8 = (lane >> 4) << 3;
  float csum[4] = {0.f, 0.f, 0.f, 0.f};
#pragma unroll
  for (int i = 0; i < 2; ++i) {
#pragma unroll
    for (int r = 0; r < 8; ++r) {
      const int rg = Abase + wr * 32 + i * 16 + half8 + r;
      float rs = 0.f;
#pragma unroll
      for (int j = 0; j < 4; ++j) {
        const int cg = Bbase + wc * 64 + j * 16 + mi;
        const float val = acc[i][j][r] * INV_T;
        if (cg - rg == BN) atomicAdd(pos_acc, val);
        const float e = (rg == cg) ? 0.0f : __expf(val);
        csum[j] += e;
        rs += e;
      }
      // row sum over this wave tile's 64 columns
      rs += __shfl_xor(rs, 1, 32);
      rs += __shfl_xor(rs, 2, 32);
      rs += __shfl_xor(rs, 4, 32);
      rs += __shfl_xor(rs, 8, 32);
      if (mi == 0) atomicAdd(&rowsum[rg], rs);
    }
  }
  // symmetry: strict-upper tiles also feed the mirrored rows (columns of this tile)
  if (upper) {
#pragma unroll
    for (int j = 0; j < 4; ++j) {
      float c = csum[j] + __shfl_xor(csum[j], 16, 32);
      if (lane < 16) atomicAdd(&rowsum[Bbase + wc * 64 + j * 16 + mi], c);
    }
  }
}

// ---------------- Kernel 3: loss = (sum log rowsum - 2*pos) / N ----------------
__global__ __launch_bounds__(256) void ntx_finalize_kernel(const float* __restrict__ rowsum,
                                                           const float* __restrict__ pos_acc,
                                                           float* __restrict__ out) {
  const int t = threadIdx.x;
  float s = 0.f;
  for (int n = t; n < NN; n += 256) s += logf(rowsum[n]);
#pragma unroll
  for (int m = 16; m >= 1; m >>= 1) s += __shfl_xor(s, m, 32);
  __shared__ float red[8];
  if ((t & 31) == 0) red[t >> 5] = s;
  __syncthreads();
  if (t == 0) {
    float tot = 0.f;
#pragma unroll
    for (int i = 0; i < 8; ++i) tot += red[i];
    out[0] = (tot - 2.0f * pos_acc[0]) / (float)NN;
  }
}

extern "C" void kernel_launch(void* const* d_in, const int* in_sizes, int n_in,
                              void* d_out, int out_size, void* d_ws, size_t ws_size,
                              hipStream_t stream) {
  const float* zi = (const float*)d_in[0];
  const float* zj = (const float*)d_in[1];
  float* inv_norm = (float*)d_ws;  // NN floats
  float* rowsum = inv_norm + NN;   // NN floats
  float* pos_acc = rowsum + NN;    // 1 float
  (void)in_sizes; (void)n_in; (void)out_size; (void)ws_size;

  ntx_norm_kernel<<<NN, 256, 0, stream>>>(zi, zj, inv_norm, rowsum, pos_acc);
  dim3 grid(NN / TILE, NN / TILE);
  ntx_gram_kernel<<<grid, 256, 0, stream>>>(zi, zj, inv_norm, rowsum, pos_acc);
  ntx_finalize_kernel<<<1, 256, 0, stream>>>(rowsum, pos_acc, (float*)d_out);
}